// SpectralStep_64201171141284
// MI455X (gfx1250) — compile-verified
//
#include <hip/hip_runtime.h>
#include <math.h>

// SpectralStep for MI455X (gfx1250, wave32).
// Bandwidth-bound pipeline (~0.95 GB HBM traffic -> ~40us @ 23.3 TB/s).
// WMMA (v_wmma_f32_16x16x4_f32) used for the pooled x conv_w GEMM.
// FFT twiddles come from a per-block LDS table (one sincosf per thread at
// block start) instead of ~600M libm sincosf calls in the butterfly loops.

typedef float v2f __attribute__((ext_vector_type(2)));
typedef float v8f __attribute__((ext_vector_type(8)));

#define PI_F 3.14159265358979323846f

__device__ __forceinline__ int brev9(int i) { return (int)(__brev((unsigned)i) >> 23); }
__device__ __forceinline__ int brev8(int i) { return (int)(__brev((unsigned)i) >> 24); }

// ---------------------------------------------------------------------------
// K2: per-row forward FFT (N=512, DIF: natural in -> bit-reversed out).
// Writes unscaled spectrum bins k<64 to XL[(bc*64+k)*256 + h] (column-major
// for the H pass) and the 64 pooled means (mean over groups of 8 along W).
// ---------------------------------------------------------------------------
__global__ __launch_bounds__(256)
void rfft_rows_kernel(const float* __restrict__ x,
                      float* __restrict__ pooled,
                      float2* __restrict__ XL)
{
    __shared__ float sre[512];
    __shared__ float sim[512];
    __shared__ float twr[256];          // W_512^t = exp(-2*pi*i*t/512), t<256
    __shared__ float twi[256];
    const int tid = threadIdx.x;
    const size_t row = blockIdx.x;                 // (b*64+c)*256 + h
    const float* xr = x + row * 512;

    sre[tid]       = xr[tid];
    sre[tid + 256] = xr[tid + 256];
    sim[tid]       = 0.f;
    sim[tid + 256] = 0.f;
    {
        float sn, cs;
        sincosf(-2.f * PI_F * (float)tid / 512.f, &sn, &cs);
        twr[tid] = cs; twi[tid] = sn;
    }
    __syncthreads();

    if (tid < 64) {
        float s = 0.f;
        #pragma unroll
        for (int j = 0; j < 8; ++j) s += sre[tid * 8 + j];
        pooled[row * 64 + tid] = s * 0.125f;       // mean of 8
    }
    __syncthreads();

    // DIF forward FFT (sign -1), 9 stages, 256 butterflies/stage.
    // Stage-m twiddle: exp(-2*pi*i*j/m) = table[j << sh], sh = log2(512/m).
    for (int m = 512, sh = 0; m >= 2; m >>= 1, ++sh) {
        const int half = m >> 1;
        const int j  = tid & (half - 1);
        const int i1 = (tid / half) * m + j;
        const int i2 = i1 + half;
        const float cs = twr[j << sh];
        const float sn = twi[j << sh];
        const float ar = sre[i1], ai = sim[i1];
        const float br = sre[i2], bi = sim[i2];
        const float tr = ar - br, ti = ai - bi;
        sre[i1] = ar + br; sim[i1] = ai + bi;
        sre[i2] = tr * cs - ti * sn;
        sim[i2] = tr * sn + ti * cs;
        __syncthreads();
    }

    if (tid < 64) {
        const size_t bc = row >> 8;
        const int h = (int)(row & 255);
        const int p = brev9(tid);                  // X[k] sits at index brev(k)
        XL[(bc * 64 + (size_t)tid) * 256 + h] = make_float2(sre[p], sim[p]);
    }
}

// ---------------------------------------------------------------------------
// K1: filter generation. One wave computes a 16(M)x64(O) params tile with
// 64 x v_wmma_f32_16x16x4_f32 (K=64 in 16 steps, 4 N-tiles), then the
// nonlinearities and the rank reduction via width-16 shuffles.
// ISA layouts (CDNA5 ISA):
//   A 16x4 f32 : lane l holds M=l%16; VGPR0 K=2*(l/16), VGPR1 K=2*(l/16)+1
//   B 4x16 f32 : lane l holds N=l%16; VGPR0 K=2*(l/16), VGPR1 K=2*(l/16)+1
//   C/D 16x16  : VGPR v: lanes 0-15 -> M=v, lanes 16-31 -> M=v+8; N=l%16
// ---------------------------------------------------------------------------
__global__ __launch_bounds__(32)
void filt_kernel(const float* __restrict__ pooled,
                 const float* __restrict__ conv_w,   // [64 o][64 c] row-major
                 const float* __restrict__ conv_b,   // [64]
                 const float* __restrict__ dt,       // [8]
                 float2* __restrict__ filt)          // [B][H*64]
{
    const int lane = threadIdx.x;
    const int lo   = lane & 15;
    const int hi   = lane >> 4;
    const int tile = blockIdx.x;                    // 0..1023 (16384/16 per b)
    const int b    = blockIdx.y;                    // 0..7
    const int m0   = tile * 16;                     // position base in (H*64)
    const float* pb = pooled + (size_t)b * (64 * 16384);  // + c*16384 + m

    v8f acc0 = {}, acc1 = {}, acc2 = {}, acc3 = {};

    #pragma unroll
    for (int s = 0; s < 16; ++s) {
        const int k0 = 4 * s + 2 * hi;
        v2f a;
        a.x = pb[(size_t)k0 * 16384 + m0 + lo];
        a.y = pb[(size_t)(k0 + 1) * 16384 + m0 + lo];

        v2f b0, b1, b2, b3;                         // B[k][o] = conv_w[o*64+k]
        b0.x = conv_w[(lo     ) * 64 + k0]; b0.y = conv_w[(lo     ) * 64 + k0 + 1];
        b1.x = conv_w[(lo + 16) * 64 + k0]; b1.y = conv_w[(lo + 16) * 64 + k0 + 1];
        b2.x = conv_w[(lo + 32) * 64 + k0]; b2.y = conv_w[(lo + 32) * 64 + k0 + 1];
        b3.x = conv_w[(lo + 48) * 64 + k0]; b3.y = conv_w[(lo + 48) * 64 + k0 + 1];

        acc0 = __builtin_amdgcn_wmma_f32_16x16x4_f32(false, a, false, b0, (short)0, acc0, false, false);
        acc1 = __builtin_amdgcn_wmma_f32_16x16x4_f32(false, a, false, b1, (short)0, acc1, false, false);
        acc2 = __builtin_amdgcn_wmma_f32_16x16x4_f32(false, a, false, b2, (short)0, acc2, false, false);
        acc3 = __builtin_amdgcn_wmma_f32_16x16x4_f32(false, a, false, b3, (short)0, acc3, false, false);
    }

    const float dtb   = dt[b];
    const float bias0 = conv_b[lo];
    const float bias1 = conv_b[lo + 16];
    const float bias2 = conv_b[lo + 32];
    const float bias3 = conv_b[lo + 48];

    float resre[8], resim[8];
    #pragma unroll
    for (int v = 0; v < 8; ++v) {
        float re = 0.f, im = 0.f;
        // rank o = lo (nu from acc0, theta from acc2)
        {
            const float pnu = acc0[v] + bias0;
            const float pth = acc2[v] + bias2;
            const float nu  = (pnu > 20.f) ? pnu : log1pf(expf(pnu));
            const float th  = tanhf(pth) * PI_F;
            const float dec = expf(-nu * dtb);
            float sn, cs; sincosf(th * dtb, &sn, &cs);
            re += dec * cs; im += dec * sn;
        }
        // rank o = lo+16 (nu from acc1, theta from acc3)
        {
            const float pnu = acc1[v] + bias1;
            const float pth = acc3[v] + bias3;
            const float nu  = (pnu > 20.f) ? pnu : log1pf(expf(pnu));
            const float th  = tanhf(pth) * PI_F;
            const float dec = expf(-nu * dtb);
            float sn, cs; sincosf(th * dtb, &sn, &cs);
            re += dec * cs; im += dec * sn;
        }
        // reduce the 16 o-values held across each 16-lane half
        re += __shfl_xor(re, 1, 16); im += __shfl_xor(im, 1, 16);
        re += __shfl_xor(re, 2, 16); im += __shfl_xor(im, 2, 16);
        re += __shfl_xor(re, 4, 16); im += __shfl_xor(im, 4, 16);
        re += __shfl_xor(re, 8, 16); im += __shfl_xor(im, 8, 16);
        resre[v] = re; resim[v] = im;
    }

    if (lo == 0) {
        float2* fb = filt + (size_t)b * 16384 + m0 + hi * 8;  // M = v + 8*hi
        #pragma unroll
        for (int v = 0; v < 8; ++v) fb[v] = make_float2(resre[v], resim[v]);
    }
}

// ---------------------------------------------------------------------------
// K3: per low-frequency column (w<64): FFT_256 (DIF) -> * filt (at brev idx)
// -> IFFT_256 (DIT, conjugate twiddles) -> * 1/256, in place on XL.
// ---------------------------------------------------------------------------
__global__ __launch_bounds__(128)
void hfilter_kernel(float2* __restrict__ XL, const float2* __restrict__ filt)
{
    __shared__ float sre[256];
    __shared__ float sim[256];
    __shared__ float twr[128];          // W_256^t, t<128
    __shared__ float twi[128];
    const int tid = threadIdx.x;                   // 0..127
    const int col = blockIdx.x;                    // bc*64 + w
    const int w   = col & 63;
    const int b   = col >> 12;                     // bc = b*64 + c
    float2* cp = XL + (size_t)col * 256;

    float2 v0 = cp[tid], v1 = cp[tid + 128];
    sre[tid] = v0.x; sim[tid] = v0.y;
    sre[tid + 128] = v1.x; sim[tid + 128] = v1.y;
    {
        float sn, cs;
        sincosf(-2.f * PI_F * (float)tid / 256.f, &sn, &cs);
        twr[tid] = cs; twi[tid] = sn;
    }
    __syncthreads();

    // forward DIF (sign -1), 8 stages, 128 butterflies/stage
    for (int m = 256, sh = 0; m >= 2; m >>= 1, ++sh) {
        const int half = m >> 1;
        const int j  = tid & (half - 1);
        const int i1 = (tid / half) * m + j;
        const int i2 = i1 + half;
        const float cs = twr[j << sh];
        const float sn = twi[j << sh];
        const float ar = sre[i1], ai = sim[i1];
        const float br = sre[i2], bi = sim[i2];
        const float tr = ar - br, ti = ai - bi;
        sre[i1] = ar + br; sim[i1] = ai + bi;
        sre[i2] = tr * cs - ti * sn;
        sim[i2] = tr * sn + ti * cs;
        __syncthreads();
    }

    // complex multiply by filt; position p holds frequency brev8(p)
    {
        const float2* fb = filt + (size_t)b * 16384;
        const float2 f0 = fb[(size_t)brev8(tid) * 64 + w];
        float r = sre[tid], i = sim[tid];
        sre[tid] = r * f0.x - i * f0.y;
        sim[tid] = r * f0.y + i * f0.x;
        const int p1 = tid + 128;
        const float2 f1 = fb[(size_t)brev8(p1) * 64 + w];
        r = sre[p1]; i = sim[p1];
        sre[p1] = r * f1.x - i * f1.y;
        sim[p1] = r * f1.y + i * f1.x;
    }
    __syncthreads();

    // inverse DIT (sign +1, conjugate twiddles), bit-reversed in -> natural out
    for (int m = 2, sh = 7; m <= 256; m <<= 1, --sh) {
        const int half = m >> 1;
        const int j  = tid & (half - 1);
        const int i1 = (tid / half) * m + j;
        const int i2 = i1 + half;
        const float cs =  twr[j << sh];
        const float sn = -twi[j << sh];
        const float br = sre[i2], bi = sim[i2];
        const float tr = br * cs - bi * sn;
        const float ti = br * sn + bi * cs;
        const float ar = sre[i1], ai = sim[i1];
        sre[i1] = ar + tr; sim[i1] = ai + ti;
        sre[i2] = ar - tr; sim[i2] = ai - ti;
        __syncthreads();
    }

    cp[tid]       = make_float2(sre[tid] * (1.f / 256.f), sim[tid] * (1.f / 256.f));
    cp[tid + 128] = make_float2(sre[tid + 128] * (1.f / 256.f), sim[tid + 128] * (1.f / 256.f));
}

// ---------------------------------------------------------------------------
// K4: per row: recompute forward DFT (DIF), splice filtered bins k<64 (and
// Hermitian mirrors 512-k) at bit-reversed positions, inverse DIT, * 1/512.
// ---------------------------------------------------------------------------
__global__ __launch_bounds__(256)
void irfft_rows_kernel(const float* __restrict__ x,
                       const float2* __restrict__ XL,
                       float* __restrict__ out)
{
    __shared__ float sre[512];
    __shared__ float sim[512];
    __shared__ float twr[256];          // W_512^t, t<256
    __shared__ float twi[256];
    const int tid = threadIdx.x;
    const size_t row = blockIdx.x;
    const float* xr = x + row * 512;

    sre[tid]       = xr[tid];
    sre[tid + 256] = xr[tid + 256];
    sim[tid]       = 0.f;
    sim[tid + 256] = 0.f;
    {
        float sn, cs;
        sincosf(-2.f * PI_F * (float)tid / 512.f, &sn, &cs);
        twr[tid] = cs; twi[tid] = sn;
    }
    __syncthreads();

    // forward DIF (sign -1)
    for (int m = 512, sh = 0; m >= 2; m >>= 1, ++sh) {
        const int half = m >> 1;
        const int j  = tid & (half - 1);
        const int i1 = (tid / half) * m + j;
        const int i2 = i1 + half;
        const float cs = twr[j << sh];
        const float sn = twi[j << sh];
        const float ar = sre[i1], ai = sim[i1];
        const float br = sre[i2], bi = sim[i2];
        const float tr = ar - br, ti = ai - bi;
        sre[i1] = ar + br; sim[i1] = ai + bi;
        sre[i2] = tr * cs - ti * sn;
        sim[i2] = tr * sn + ti * cs;
        __syncthreads();
    }

    // splice in the H-filtered low-frequency bins
    if (tid < 64) {
        const size_t bc = row >> 8;
        const int h = (int)(row & 255);
        const float2 v = XL[(bc * 64 + (size_t)tid) * 256 + h];
        sre[brev9(tid)] = v.x;
        sim[brev9(tid)] = v.y;
        if (tid > 0) {                              // Hermitian mirror bin 512-k
            const int p2 = brev9(512 - tid);
            sre[p2] = v.x;
            sim[p2] = -v.y;
        }
    }
    __syncthreads();

    // inverse DIT (sign +1, conjugate twiddles), natural-order output
    for (int m = 2, sh = 8; m <= 512; m <<= 1, --sh) {
        const int half = m >> 1;
        const int j  = tid & (half - 1);
        const int i1 = (tid / half) * m + j;
        const int i2 = i1 + half;
        const float cs =  twr[j << sh];
        const float sn = -twi[j << sh];
        const float br = sre[i2], bi = sim[i2];
        const float tr = br * cs - bi * sn;
        const float ti = br * sn + bi * cs;
        const float ar = sre[i1], ai = sim[i1];
        sre[i1] = ar + tr; sim[i1] = ai + ti;
        sre[i2] = ar - tr; sim[i2] = ai - ti;
        __syncthreads();
    }

    out[row * 512 + tid]       = sre[tid]       * (1.f / 512.f);
    out[row * 512 + tid + 256] = sre[tid + 256] * (1.f / 512.f);
}

// ---------------------------------------------------------------------------
extern "C" void kernel_launch(void* const* d_in, const int* in_sizes, int n_in,
                              void* d_out, int out_size, void* d_ws, size_t ws_size,
                              hipStream_t stream)
{
    const float* x      = (const float*)d_in[0];  // (8,64,256,512)
    const float* dt     = (const float*)d_in[1];  // (8,)
    const float* conv_w = (const float*)d_in[2];  // (64,64)
    const float* conv_b = (const float*)d_in[3];  // (64,)
    float* out = (float*)d_out;                   // (8,64,256,512)

    char* ws = (char*)d_ws;
    float*  pooled = (float*)ws;                               // 32 MB: (B,C,H,64) f32
    float2* XL     = (float2*)(ws + (size_t)32 * 1024 * 1024); // 64 MB: (B,C,64,H) c64
    float2* filt   = (float2*)(ws + (size_t)96 * 1024 * 1024); //  1 MB: (B,H,64)  c64

    rfft_rows_kernel <<<dim3(131072),  dim3(256), 0, stream>>>(x, pooled, XL);
    filt_kernel      <<<dim3(1024, 8), dim3(32),  0, stream>>>(pooled, conv_w, conv_b, dt, filt);
    hfilter_kernel   <<<dim3(32768),   dim3(128), 0, stream>>>(XL, filt);
    irfft_rows_kernel<<<dim3(131072),  dim3(256), 0, stream>>>(x, XL, out);

    (void)in_sizes; (void)n_in; (void)out_size; (void)ws_size;
}